// QCQP_cvxpy_14740327760511
// MI455X (gfx1250) — compile-verified
//
#include <hip/hip_runtime.h>
#include <hip/hip_bf16.h>

typedef __attribute__((ext_vector_type(2))) float v2f;
typedef __attribute__((ext_vector_type(8))) float v8f;
typedef __attribute__((ext_vector_type(4))) unsigned int u32x4;
typedef __attribute__((ext_vector_type(8))) unsigned int u32x8;

#define BATCH 512
#define NDIM 256
#define NCONE 128
#define MAX_ITER 100
#define POWER_ITERS 20
#define HSTRIDE 260               // 256 + 4 pad DWORDs inserted by TDM padding
#define SOLVE_LDS_BYTES ((NDIM * HSTRIDE + 2 * NDIM) * 4)   // 268,288 B < 320 KB

// ---------------------------------------------------------------------------
// Kernel 1: H[b] = P[b]^T P[b]  (256x256x256 f32 GEMM per batch)
// Block computes a 64x128 tile of H with 8 waves arranged 2x4; each wave owns
// a 32x32 strip = 2x2 grid of 16x16 f32 accumulators (V_WMMA_F32_16X16X4).
// ---------------------------------------------------------------------------
__global__ __launch_bounds__(256) void gram_kernel(const float* __restrict__ P,
                                                   float* __restrict__ H) {
  const int b   = blockIdx.z;
  const int ti0 = blockIdx.y * 64;
  const int tj0 = blockIdx.x * 128;
  const int tid  = threadIdx.x;
  const int lane = tid & 31;
  const int wave = tid >> 5;
  const int wm   = wave >> 2;
  const int wn   = wave & 3;
  const int half = lane >> 4;
  const int lr   = lane & 15;

  __shared__ float As[16][64];
  __shared__ float Bs[16][128];

  const float* Pb = P + (size_t)b * NDIM * NDIM;
  float*       Hb = H + (size_t)b * NDIM * NDIM;

  v8f acc00 = {}, acc01 = {}, acc10 = {}, acc11 = {};

  const int kkA = tid >> 4;
  const int cA  = (tid & 15) * 4;
  const int kkB = tid >> 5;
  const int cB  = (tid & 31) * 4;

  for (int k0 = 0; k0 < NDIM; k0 += 16) {
    const float4 av  = *(const float4*)(Pb + (size_t)(k0 + kkA) * NDIM + ti0 + cA);
    const float4 bv0 = *(const float4*)(Pb + (size_t)(k0 + kkB) * NDIM + tj0 + cB);
    const float4 bv1 = *(const float4*)(Pb + (size_t)(k0 + kkB + 8) * NDIM + tj0 + cB);
    __syncthreads();
    *(float4*)&As[kkA][cA]     = av;
    *(float4*)&Bs[kkB][cB]     = bv0;
    *(float4*)&Bs[kkB + 8][cB] = bv1;
    __syncthreads();
#pragma unroll
    for (int kc = 0; kc < 16; kc += 4) {
      v2f a0, a1, b0, b1;
      a0.x = As[kc + 2 * half + 0][wm * 32 + lr];
      a0.y = As[kc + 2 * half + 1][wm * 32 + lr];
      a1.x = As[kc + 2 * half + 0][wm * 32 + 16 + lr];
      a1.y = As[kc + 2 * half + 1][wm * 32 + 16 + lr];
      b0.x = Bs[kc + 2 * half + 0][wn * 32 + lr];
      b0.y = Bs[kc + 2 * half + 1][wn * 32 + lr];
      b1.x = Bs[kc + 2 * half + 0][wn * 32 + 16 + lr];
      b1.y = Bs[kc + 2 * half + 1][wn * 32 + 16 + lr];
      acc00 = __builtin_amdgcn_wmma_f32_16x16x4_f32(false, a0, false, b0,
                                                    (short)0, acc00, false, false);
      acc01 = __builtin_amdgcn_wmma_f32_16x16x4_f32(false, a0, false, b1,
                                                    (short)0, acc01, false, false);
      acc10 = __builtin_amdgcn_wmma_f32_16x16x4_f32(false, a1, false, b0,
                                                    (short)0, acc10, false, false);
      acc11 = __builtin_amdgcn_wmma_f32_16x16x4_f32(false, a1, false, b1,
                                                    (short)0, acc11, false, false);
    }
  }

  const int row0 = ti0 + wm * 32 + half * 8;
  const int col0 = tj0 + wn * 32 + lr;
#pragma unroll
  for (int r = 0; r < 8; ++r) {
    Hb[(size_t)(row0 + r) * NDIM + col0]           = acc00[r];
    Hb[(size_t)(row0 + r) * NDIM + col0 + 16]      = acc01[r];
    Hb[(size_t)(row0 + 16 + r) * NDIM + col0]      = acc10[r];
    Hb[(size_t)(row0 + 16 + r) * NDIM + col0 + 16] = acc11[r];
  }
}

// ---------------------------------------------------------------------------
// Kernel 2 (fused power-iteration + FISTA), one block per batch.
// H[b] (256KB) is DMA'd into LDS ONCE via the Tensor Data Mover with LDS
// padding (4 pad DWORDs every 256 -> row stride 260, bank-spread), then all
// 121 matvec sweeps run out of LDS. Each of the 8 waves issues one
// tensor_load_to_lds for its 32-row slice, then s_wait_tensorcnt + barrier.
// ---------------------------------------------------------------------------
__global__ __launch_bounds__(256) void solve_kernel(const float* __restrict__ H,
                                                    const float* __restrict__ q,
                                                    const float* __restrict__ l_n,
                                                    const float* __restrict__ mu,
                                                    float* __restrict__ out) {
  extern __shared__ __align__(16) float smem[];
  float* Hs = smem;                       // [256][260] padded matrix
  float* va = smem + NDIM * HSTRIDE;      // v (power) / y (FISTA)
  float* rb = va + NDIM;                  // reduction / z (FISTA)

  const int b = blockIdx.x;
  const int i = threadIdx.x;

  // ---- TDM: load 32 rows of H[b] per wave into padded LDS ----------------
  {
    const unsigned wv = (unsigned)__builtin_amdgcn_readfirstlane((int)(threadIdx.x >> 5));
    const unsigned long long gaddr =
        (unsigned long long)(const void*)(H + (size_t)b * NDIM * NDIM +
                                          (size_t)wv * 32 * NDIM);
    const unsigned lds_off = wv * 32u * HSTRIDE * 4u;   // byte offset in LDS

    // D# group 0: count=1 | lds_addr | global_addr[56:0] | type=2
    const u32x4 g0 = { 1u,
                       lds_off,
                       (unsigned)gaddr,
                       (unsigned)((gaddr >> 32) & 0x1FFFFFFu) | (2u << 30) };
    // D# group 1: data_size=4B, pad_enable, pad_interval=7 (256 dw),
    // pad_amount=3 (4 dw); tensor_dim0=256, tensor_dim1=32, tile=256x32,
    // tensor_dim0_stride=256.
    const u32x8 g1 = { (2u << 16) | (1u << 20) | (7u << 22) | (3u << 25),
                       256u << 16,     // tensor_dim0[15:0] @ bits 63:48
                       32u << 16,      // tensor_dim1[15:0] @ bits 95:80
                       256u << 16,     // tile_dim0 @ bits 127:112
                       32u,            // tile_dim1
                       256u,           // tensor_dim0_stride[31:0]
                       0u, 0u };
    asm volatile("tensor_load_to_lds %0, %1, null, null"
                 :: "s"(g0), "s"(g1)
                 : "memory");
    __builtin_amdgcn_s_wait_tensorcnt(0);
  }
  __syncthreads();

  const float4* H4 = (const float4*)(Hs + (size_t)i * HSTRIDE);  // 16B aligned

  // ---- power iteration ----------------------------------------------------
  va[i] = 1.0f;
  __syncthreads();

  for (int it = 0; it < POWER_ITERS; ++it) {
    float w = 0.0f;
#pragma unroll 4
    for (int j = 0; j < NDIM / 4; ++j) {
      const float4 h = H4[j];
      w += h.x * va[4 * j] + h.y * va[4 * j + 1] + h.z * va[4 * j + 2] + h.w * va[4 * j + 3];
    }
    rb[i] = w * w;
    __syncthreads();
    for (int s = 128; s > 0; s >>= 1) {
      if (i < s) rb[i] += rb[i + s];
      __syncthreads();
    }
    const float nrm = sqrtf(rb[0]);
    __syncthreads();
    va[i] = w / (nrm + 1e-12f);
    __syncthreads();
  }

  // Rayleigh quotient L = (v.Hv)/(v.v)
  float hv = 0.0f;
#pragma unroll 4
  for (int j = 0; j < NDIM / 4; ++j) {
    const float4 h = H4[j];
    hv += h.x * va[4 * j] + h.y * va[4 * j + 1] + h.z * va[4 * j + 2] + h.w * va[4 * j + 3];
  }
  rb[i] = va[i] * hv;
  __syncthreads();
  for (int s = 128; s > 0; s >>= 1) {
    if (i < s) rb[i] += rb[i + s];
    __syncthreads();
  }
  const float num = rb[0];
  __syncthreads();
  rb[i] = va[i] * va[i];
  __syncthreads();
  for (int s = 128; s > 0; s >>= 1) {
    if (i < s) rb[i] += rb[i + s];
    __syncthreads();
  }
  const float den = rb[0];
  __syncthreads();
  const float st = 1.0f / (1.05f * fmaxf(num / den, 1e-12f));  // uniform

  // ---- FISTA (va = y, rb = z) --------------------------------------------
  const float bq = q[b * NDIM + i];
  const int   p  = i >> 1;
  const float r  = mu[b * NCONE + p] * l_n[b * NCONE + p];

  float li = 0.0f;   // l0 = project(0) = 0
  float t  = 1.0f;
  va[i] = 0.0f;
  __syncthreads();

  for (int it = 0; it < MAX_ITER; ++it) {
    float g = bq;
#pragma unroll 4
    for (int j = 0; j < NDIM / 4; ++j) {
      const float4 h = H4[j];
      g += h.x * va[4 * j] + h.y * va[4 * j + 1] + h.z * va[4 * j + 2] + h.w * va[4 * j + 3];
    }
    const float zi = va[i] - st * g;
    rb[i] = zi;
    __syncthreads();                       // matvec reads of y done; z visible
    const float z0 = rb[i & ~1];
    const float z1 = rb[i | 1];
    const float nrm   = sqrtf(z0 * z0 + z1 * z1);
    const float scale = (nrm > r) ? (r / fmaxf(nrm, 1e-30f)) : 1.0f;
    const float lnew  = zi * scale;
    const float tnew  = 0.5f * (1.0f + sqrtf(1.0f + 4.0f * t * t));
    const float ynew  = lnew + ((t - 1.0f) / tnew) * (lnew - li);
    li = lnew;
    t  = tnew;
    __syncthreads();                       // z reads done before y overwrite
    va[i] = ynew;
    __syncthreads();
  }

  out[b * NDIM + i] = li;
}

// ---------------------------------------------------------------------------
extern "C" void kernel_launch(void* const* d_in, const int* in_sizes, int n_in,
                              void* d_out, int out_size, void* d_ws, size_t ws_size,
                              hipStream_t stream) {
  (void)in_sizes; (void)n_in; (void)out_size; (void)ws_size;
  const float* P   = (const float*)d_in[0];
  const float* q   = (const float*)d_in[1];
  const float* l_n = (const float*)d_in[2];
  const float* mu  = (const float*)d_in[3];
  float* out = (float*)d_out;

  float* H = (float*)d_ws;                        // 512*256*256 f32 = 128MB

  // allow >64KB dynamic LDS (262KB: whole H[b] resident per WGP)
  (void)hipFuncSetAttribute((const void*)solve_kernel,
                            hipFuncAttributeMaxDynamicSharedMemorySize,
                            SOLVE_LDS_BYTES);

  dim3 g1(NDIM / 128, NDIM / 64, BATCH);          // 2 x 4 x 512 tiles
  gram_kernel<<<g1, 256, 0, stream>>>(P, H);
  solve_kernel<<<BATCH, 256, SOLVE_LDS_BYTES, stream>>>(H, q, l_n, mu, out);
}